// VectorQuantizer_13030930776476
// MI455X (gfx1250) — compile-verified
//
#include <hip/hip_runtime.h>
#include <hip/hip_bf16.h>
#include <math.h>

typedef __attribute__((ext_vector_type(2))) float v2f;
typedef __attribute__((ext_vector_type(4))) float v4f;
typedef __attribute__((ext_vector_type(8))) float v8f;
typedef float v4f_u __attribute__((ext_vector_type(4), aligned(4)));

#define D_DIM 64
#define K_DIM 512
#define N_ROWS 131072              // 32*64*64 rows of length 64
#define NUMEL  8388608             // N_ROWS * D_DIM
#define ROWS_PER_WAVE 32           // two 16-row WMMA tiles per wave
#define WAVES_PER_BLOCK 8
#define ROWS_PER_BLOCK (ROWS_PER_WAVE * WAVES_PER_BLOCK)   // 256
#define NUM_BLOCKS (N_ROWS / ROWS_PER_BLOCK)               // 512
#define NUM_WAVES  (N_ROWS / ROWS_PER_WAVE)                // 4096

// workspace layout (4-byte units):
//   [0, 32768)          : float ET[512][64]   transposed codebook (16B aligned)
//   [32768, 33280)      : unsigned hist[512]
//   [33280, 33792)      : float halfc2[512]   (0.5 * ||e_k||^2)
//   [33792, 33792+4096) : float ssePartial[NUM_WAVES]
#define WS_ET      0
#define WS_HIST    32768
#define WS_HALFC2  33280
#define WS_PARTIAL 33792

__global__ __launch_bounds__(512)
void vq_init(const float* __restrict__ E, float* __restrict__ ET,
             float* __restrict__ halfc2, unsigned* __restrict__ hist) {
    const int k = threadIdx.x;                 // 512 threads, one per code
    float s = 0.0f;
#pragma unroll
    for (int d = 0; d < D_DIM; ++d) {
        const float e = E[d * K_DIM + k];      // coalesced read across threads
        ET[k * D_DIM + d] = e;                 // contiguous per-thread write
        s += e * e;
    }
    halfc2[k] = 0.5f * s;
    hist[k] = 0u;
}

__global__ __launch_bounds__(256)
void vq_main(const float* __restrict__ X, const float* __restrict__ ET,
             const float* __restrict__ halfc2, unsigned* __restrict__ hist,
             float* __restrict__ partial, float* __restrict__ qout,
             float* __restrict__ iout) {
    const int lane = threadIdx.x & 31;
    const int wave = threadIdx.x >> 5;
    const int waveGlobal = blockIdx.x * WAVES_PER_BLOCK + wave;
    const int row_base = waveGlobal * ROWS_PER_WAVE;
    const int mrow = lane & 15;       // M index within 16-row tile
    const int sel  = lane >> 4;       // which K-pair this half-wave holds

    // ---- two A tiles: 32 rows x 64 deep, 16 chunks of 16x4 fp32 each ----
    // ISA layout: lanes 0-15 hold (K=0,K=1), lanes 16-31 hold (K=2,K=3), M = lane%16
    v2f a0[16], a1[16];
    {
        const float* ar0 = X + (size_t)(row_base + mrow) * D_DIM + sel * 2;
        const float* ar1 = ar0 + 16 * D_DIM;
#pragma unroll
        for (int c = 0; c < 16; ++c) {
            a0[c] = *(const v2f*)(ar0 + c * 4);
            a1[c] = *(const v2f*)(ar1 + c * 4);
        }
    }

    float best[2][8];
    int   bidx[2][8];
#pragma unroll
    for (int t = 0; t < 2; ++t)
#pragma unroll
        for (int r = 0; r < 8; ++r) { best[t][r] = -3.4e38f; bidx[t][r] = 0; }

    // ---- sweep the 512 codes in 16-wide column tiles ----
    for (int j = 0; j < K_DIM; j += 16) {
        const int col = j + mrow;
        const float hc = halfc2[col];
        // B from transposed codebook: rows (4c+2sel, 4c+2sel+1) are adjacent
        const float* bp = ET + (size_t)col * D_DIM + sel * 2;
        const float nh = -hc;                 // fold -0.5||e||^2 into C init
        v8f acc0 = {nh, nh, nh, nh, nh, nh, nh, nh};
        v8f acc1 = acc0;
#pragma unroll
        for (int c = 0; c < 16; ++c) {
            const v2f b = *(const v2f*)(bp + c * 4);   // one b64, feeds 2 WMMAs
            acc0 = __builtin_amdgcn_wmma_f32_16x16x4_f32(
                false, a0[c], false, b, (short)0, acc0, false, false);
            acc1 = __builtin_amdgcn_wmma_f32_16x16x4_f32(
                false, a1[c], false, b, (short)0, acc1, false, false);
        }
        // score = f.e - 0.5||e||^2 ; argmax == argmin distance.
        // strict '>' keeps earliest (lowest) column on exact ties.
#pragma unroll
        for (int r = 0; r < 8; ++r) {
            if (acc0[r] > best[0][r]) { best[0][r] = acc0[r]; bidx[0][r] = col; }
            if (acc1[r] > best[1][r]) { best[1][r] = acc1[r]; bidx[1][r] = col; }
        }
    }

    // ---- cross-lane argmax within each 16-lane half (tie -> lower index) ----
#pragma unroll
    for (int m = 1; m <= 8; m <<= 1) {
#pragma unroll
        for (int t = 0; t < 2; ++t)
#pragma unroll
            for (int r = 0; r < 8; ++r) {
                float ov = __shfl_xor(best[t][r], m, 32);
                int   oi = __shfl_xor(bidx[t][r], m, 32);
                if (ov > best[t][r] || (ov == best[t][r] && oi < bidx[t][r])) {
                    best[t][r] = ov; bidx[t][r] = oi;
                }
            }
    }
    // tile t rows 16t+0..7 uniform across lanes 0-15 (reg r), 16t+8..15 across 16-31

    // ---- indices (as float) + histogram (integer atomics: deterministic) ----
#pragma unroll
    for (int t = 0; t < 2; ++t)
#pragma unroll
        for (int r = 0; r < 8; ++r) {
            if (lane == 0) {
                iout[row_base + 16 * t + r] = (float)bidx[t][r];
                atomicAdd(&hist[bidx[t][r]], 1u);
            }
            if (lane == 16) {
                iout[row_base + 16 * t + 8 + r] = (float)bidx[t][r];
                atomicAdd(&hist[bidx[t][r]], 1u);
            }
        }

    // ---- gather quantized rows + squared error (b128 loads, coalesced) ----
    float serr = 0.0f;
#pragma unroll
    for (int t = 0; t < 2; ++t) {
#pragma unroll
        for (int mp = 0; mp < 8; ++mp) {
            // lanes 0-15 do row 2mp, lanes 16-31 do row 2mp+1 (16 lanes/row)
            const int rA = 2 * mp, rB = 2 * mp + 1;
            const int ivA = __shfl(bidx[t][rA & 7], (rA >> 3) * 16, 32);
            const int ivB = __shfl(bidx[t][rB & 7], (rB >> 3) * 16, 32);
            const int idx_lane = (lane < 16) ? ivA : ivB;
            const int d0 = (lane & 15) * 4;
            const int m = 2 * mp + (lane >> 4);
            const v4f e = *(const v4f*)(ET + (size_t)idx_lane * D_DIM + d0);
            const size_t off = (size_t)(row_base + 16 * t + m) * D_DIM + d0;
            const v4f xv = *(const v4f*)(X + off);
            *(v4f_u*)(qout + off) = e;          // qout is only 4B aligned
            const v4f df = e - xv;
            serr += df.x * df.x + df.y * df.y + df.z * df.z + df.w * df.w;
        }
    }
    // wave-level reduction, then deterministic per-wave partial
#pragma unroll
    for (int m = 1; m < 32; m <<= 1) serr += __shfl_xor(serr, m, 32);
    if (lane == 0) partial[waveGlobal] = serr;
}

__global__ __launch_bounds__(512)
void vq_finalize(const unsigned* __restrict__ hist,
                 const float* __restrict__ partial,
                 float* __restrict__ out_loss, float* __restrict__ out_perp) {
    __shared__ float red[512];
    const int t = threadIdx.x;

    // deterministic fixed-order SSE reduction over 4096 wave partials
    float s = 0.0f;
#pragma unroll
    for (int i = 0; i < NUM_WAVES / 512; ++i) s += partial[t * (NUM_WAVES / 512) + i];
    red[t] = s;
    __syncthreads();
    for (int k = 256; k > 0; k >>= 1) {
        if (t < k) red[t] += red[t + k];
        __syncthreads();
    }
    const float sse = red[0];
    __syncthreads();

    // perplexity = exp(-sum p*log(p+1e-10)), p = hist/N
    const float p = (float)hist[t] * (1.0f / (float)N_ROWS);
    red[t] = p * logf(p + 1e-10f);
    __syncthreads();
    for (int k = 256; k > 0; k >>= 1) {
        if (t < k) red[t] += red[t + k];
        __syncthreads();
    }
    if (t == 0) {
        // q_latent_loss == e_latent_loss numerically => loss = 1.25 * MSE
        out_loss[0] = 1.25f * (sse / (float)NUMEL);
        out_perp[0] = expf(-red[0]);
    }
}

extern "C" void kernel_launch(void* const* d_in, const int* in_sizes, int n_in,
                              void* d_out, int out_size, void* d_ws, size_t ws_size,
                              hipStream_t stream) {
    const float* X = (const float*)d_in[0];   // (32,64,64,64) fp32
    const float* E = (const float*)d_in[1];   // (64,512) fp32
    float* out  = (float*)d_out;
    float* loss = out;                        // [0]
    float* qout = out + 1;                    // [1 .. 1+NUMEL)
    float* perp = out + 1 + NUMEL;            // [1+NUMEL]
    float* iout = out + 2 + NUMEL;            // [2+NUMEL .. 2+NUMEL+N_ROWS)

    float*    ET      = (float*)d_ws + WS_ET;
    unsigned* hist    = (unsigned*)d_ws + WS_HIST;
    float*    halfc2  = (float*)d_ws + WS_HALFC2;
    float*    partial = (float*)d_ws + WS_PARTIAL;

    vq_init<<<1, 512, 0, stream>>>(E, ET, halfc2, hist);
    vq_main<<<NUM_BLOCKS, 256, 0, stream>>>(X, ET, halfc2, hist, partial, qout, iout);
    vq_finalize<<<1, 512, 0, stream>>>(hist, partial, loss, perp);
}